// Presence_NN_sim_merger_46986942218268
// MI455X (gfx1250) — compile-verified
//
#include <hip/hip_runtime.h>
#include <math.h>

typedef __attribute__((ext_vector_type(2))) float v2f;
typedef __attribute__((ext_vector_type(8))) float v8f;
typedef int v4i_t __attribute__((__vector_size__(4 * sizeof(int))));

namespace {
constexpr int S  = 9;
constexpr int D  = 64;
constexpr int SD = S * D;          // 576 floats per batch tile
constexpr int WAVES = 4;           // 128 threads / wave32 (LDS-budget bound)
constexpr float THR = 0.9f;
constexpr float EPS = 1e-8f;
}

__global__ __launch_bounds__(128)
void slot_merger_kernel(const float* __restrict__ g_slots,
                        float* __restrict__ g_final,
                        float* __restrict__ g_mask)
{
    // Per-wave LDS working set. ls_GN: first 256 floats = 16x16 Gram (phases A-C),
    // then reused whole (16x64) for new_slots (phase E onwards).
    __shared__ __align__(16) float ls_slots[WAVES][16 * D]; // zero-padded 16x64 tile
    __shared__ __align__(16) float ls_GN[WAVES][16 * D];    // Gram alias new_slots
    __shared__ __align__(16) float ls_M [WAVES][256];       // 16x16 maskf (padded)
    __shared__ float ls_norm[WAVES][16];
    __shared__ float ls_nums[WAVES][16];
    __shared__ int   ls_first[WAVES][16];
    __shared__ int   ls_multi[WAVES][16];
    __shared__ int   ls_map  [WAVES][16];

    const int tid  = threadIdx.x;
    const int wave = tid >> 5;
    const int lane = tid & 31;
    const int row  = lane & 15;   // M (or N) index within the 16-wide tile
    const int half = lane >> 4;   // K sub-chunk selector for f32 WMMA operands

    const long long b = (long long)blockIdx.x * WAVES + wave;
    const float* gbase = g_slots + b * SD;

    float* sl  = ls_slots[wave];
    float* G   = ls_GN[wave];     // Gram view (256 floats used)
    float* NEW = ls_GN[wave];     // new_slots view (1024 floats), after Gram is dead
    float* M   = ls_M[wave];

    // ---- Stage: zero-pad rows 9..15, then one coalesced pass over the tile ----
    {
        float4* l4 = (float4*)sl;
        const float4 z4 = {0.f, 0.f, 0.f, 0.f};
        // Zero float4 indices 144..255 (rows 9..15), disjoint from real data 0..143
        #pragma unroll
        for (int i = 0; i < 3; ++i) l4[144 + lane + 32 * i] = z4;
        if (lane < 16)              l4[240 + lane]          = z4;

        const float4* g4 = (const float4*)gbase;
#if __has_builtin(__builtin_amdgcn_global_load_async_to_lds_b128)
        // CDNA5 async copy: HBM -> LDS without VGPR round-trip (ASYNCcnt-tracked)
        typedef __attribute__((address_space(1))) v4i_t* gv4p;
        typedef __attribute__((address_space(3))) v4i_t* lv4p;
        #pragma unroll
        for (int i = 0; i < 4; ++i) {
            const int idx = lane + 32 * i;
            __builtin_amdgcn_global_load_async_to_lds_b128(
                (gv4p)(g4 + idx), (lv4p)(l4 + idx), 0, 0);
        }
        if (lane < 16) {
            const int idx = 128 + lane;
            __builtin_amdgcn_global_load_async_to_lds_b128(
                (gv4p)(g4 + idx), (lv4p)(l4 + idx), 0, 0);
        }
  #if __has_builtin(__builtin_amdgcn_s_wait_asynccnt)
        __builtin_amdgcn_s_wait_asynccnt(0);
  #else
        asm volatile("s_wait_asynccnt 0" ::: "memory");
  #endif
#else
        #pragma unroll
        for (int i = 0; i < 4; ++i) l4[lane + 32 * i] = g4[lane + 32 * i];
        if (lane < 16)              l4[128 + lane]    = g4[128 + lane];
#endif
    }
    __syncthreads();

    // ---- Phase A: Gram G = slotsPad (16x64) * slotsPad^T, exact fp32 WMMA ----
    // A-operand (16x4) and B-operand (4x16 of A^T) occupy identical registers.
    // Preload all operand pairs so the 16 WMMAs issue back-to-back.
    {
        v2f amat[16];
        #pragma unroll
        for (int kc = 0; kc < 16; ++kc)
            amat[kc] = *(const v2f*)(sl + row * D + kc * 4 + half * 2);

        v8f acc = {0.f, 0.f, 0.f, 0.f, 0.f, 0.f, 0.f, 0.f};
        #pragma unroll
        for (int kc = 0; kc < 16; ++kc)
            acc = __builtin_amdgcn_wmma_f32_16x16x4_f32(
                false, amat[kc], false, amat[kc], (short)0, acc, false, false);

        #pragma unroll
        for (int i = 0; i < 8; ++i) G[(half * 8 + i) * 16 + row] = acc[i];
    }
    __syncthreads();

    // ---- Phase B: row norms from the diagonal ----
    if (lane < 16) ls_norm[wave][lane] = sqrtf(G[lane * 16 + lane]);
    __syncthreads();

    // ---- Phase C: maskf[r][c] = (G/(|r||c|+eps) > THR), zero-padded to 16x16 ----
    {
        const float* nrm = ls_norm[wave];
        #pragma unroll
        for (int t = 0; t < 8; ++t) {
            const int e = lane * 8 + t;
            const int r = e >> 4, c = e & 15;
            float v = 0.f;
            if (r < S && c < S) {
                const float sim = G[e] / (nrm[r] * nrm[c] + EPS);
                v = (sim > THR) ? 1.f : 0.f;
            }
            M[e] = v;
        }
    }
    __syncthreads();

    // ---- Phase D: per-row nums / first_idx / multi / scatter-map ----
    if (lane < 16) {
        const int s = lane;
        float cnt = 0.f; int first = 0; bool found = false;
        if (s < S) {
            for (int j = 0; j < S; ++j) {
                if (M[s * 16 + j] != 0.f) {
                    cnt += 1.f;
                    if (!found) { first = j; found = true; }
                }
            }
        }
        const int mult = (cnt > 1.f) ? 1 : 0;
        ls_nums[wave][s]  = cnt;
        ls_first[wave][s] = first;
        ls_multi[wave][s] = mult;
        ls_map[wave][s]   = mult ? first : s;   // target row for slot s
    }
    __syncthreads();

    // ---- Phase E: NEW = maskf(16x16) @ slotsPad(16x64) via WMMA ----
    // Rows 9..15 of both operands are exact zeros; no clamps, no divergence.
    #pragma unroll
    for (int tile = 0; tile < 4; ++tile) {
        v2f am[4], bm[4];
        #pragma unroll
        for (int kc = 0; kc < 4; ++kc) {
            const int k0 = kc * 4 + half * 2;
            am[kc] = *(const v2f*)(M + row * 16 + k0);
            v2f t; t.x = sl[k0 * D + tile * 16 + row];
                   t.y = sl[(k0 + 1) * D + tile * 16 + row];
            bm[kc] = t;
        }
        v8f acc2 = {0.f, 0.f, 0.f, 0.f, 0.f, 0.f, 0.f, 0.f};
        #pragma unroll
        for (int kc = 0; kc < 4; ++kc)
            acc2 = __builtin_amdgcn_wmma_f32_16x16x4_f32(
                false, am[kc], false, bm[kc], (short)0, acc2, false, false);
        #pragma unroll
        for (int i = 0; i < 8; ++i)
            NEW[(half * 8 + i) * D + tile * 16 + row] = acc2[i];
    }
    __syncthreads();

    // ---- Phase F: slot_mask[j] = (nums[j] > 0) && no zero_hit in column j ----
    if (lane < S) {
        const int j = lane;
        bool zero_hit = false;
        for (int s = 0; s < S; ++s) {
            if (M[s * 16 + j] != 0.f && j != ls_first[wave][s] && ls_multi[wave][s])
                zero_hit = true;
        }
        const bool keep = (ls_nums[wave][j] > 0.f) && !zero_hit;
        g_mask[b * S + j] = keep ? 1.f : 0.f;
    }

    // ---- Phase G: sequential-scatter semantics, emitted as coalesced b128 ----
    {
        int mp[S];
        #pragma unroll
        for (int s = 0; s < S; ++s) mp[s] = ls_map[wave][s];
        int win[S];
        #pragma unroll
        for (int t = 0; t < S; ++t) {
            int w = -1;
            #pragma unroll
            for (int s = 0; s < S; ++s) if (mp[s] == t) w = s;   // max s wins
            win[t] = w;
        }

        auto emit = [&](int idx) {   // idx indexes float4s of the 9x64 output tile
            const int t  = idx >> 4;
            const int c0 = (idx & 15) * 4;
            const int w  = win[t];
            float4 v = {0.f, 0.f, 0.f, 0.f};
            if (w >= 0) {
                if (ls_multi[wave][w]) {
                    const float inv = 1.f / (ls_nums[wave][w] + EPS);
                    const float4 n4 = *(const float4*)(NEW + w * D + c0);
                    v.x = n4.x * inv; v.y = n4.y * inv;
                    v.z = n4.z * inv; v.w = n4.w * inv;
                } else {
                    v = *(const float4*)(sl + w * D + c0);
                }
            }
            *(float4*)(g_final + b * SD + idx * 4) = v;
        };
        #pragma unroll
        for (int i = 0; i < 4; ++i) emit(lane + 32 * i);
        if (lane < 16)              emit(128 + lane);
    }
}

extern "C" void kernel_launch(void* const* d_in, const int* in_sizes, int n_in,
                              void* d_out, int out_size, void* d_ws, size_t ws_size,
                              hipStream_t stream) {
    (void)n_in; (void)out_size; (void)d_ws; (void)ws_size;
    const float* slots = (const float*)d_in[0];
    const int Btot = in_sizes[0] / SD;        // 131072
    float* out_final = (float*)d_out;
    float* out_mask  = out_final + (size_t)Btot * SD;

    const int blocks = Btot / WAVES;          // one batch per wave, 4 waves/block
    slot_merger_kernel<<<blocks, 128, 0, stream>>>(slots, out_final, out_mask);
}